// GraphProp_68908455297282
// MI455X (gfx1250) — compile-verified
//
#include <hip/hip_runtime.h>
#include <hip/hip_bf16.h>

// ---------------------------------------------------------------------------
// GraphProp GRU message passing for MI455X (gfx1250, wave32).
//
//   hv' = GRU( Linear(concat[hv, mean(hv[src]), mean(he)]) , hv )  x T rounds
//
// GEMMs run on V_WMMA_F32_16X16X4_F32 (full fp32). Each wave owns TWO 16-row
// node tiles so every weight (B) fragment loaded from L2 feeds two WMMAs.
// The hv tiles are staged into LDS by the Tensor Data Mover
// (tensor_load_to_lds + s_wait_tensorcnt). GRU gates r/z/n are combined
// entirely in registers. Edge reductions use global_atomic_add_f32.
// ---------------------------------------------------------------------------

typedef __attribute__((ext_vector_type(2))) float v2f;
typedef __attribute__((ext_vector_type(8))) float v8f;
typedef __attribute__((ext_vector_type(4))) unsigned int v4u;
typedef __attribute__((ext_vector_type(8))) unsigned int v8u;

#define H 128          // hidden size (fixed by the reference)
#define WAVE 32
#define MT 2           // row tiles (of 16 nodes) per wave

// -------------------------- small utility kernels --------------------------

__global__ void zero_kernel(float* __restrict__ p, long long n) {
    long long i = (long long)blockIdx.x * blockDim.x + threadIdx.x;
    if (i < n) p[i] = 0.0f;
}

// One thread per (edge, float4 chunk of H): accumulate he into sum_he[dst]
// and (chunk 0 only) the in-degree.
__global__ void deg_he_kernel(const float* __restrict__ he,
                              const int* __restrict__ dst,
                              float* __restrict__ degf,
                              float* __restrict__ sum_he,
                              int E) {
    long long tid = (long long)blockIdx.x * blockDim.x + threadIdx.x;
    if (tid >= (long long)E * (H / 4)) return;
    int e = (int)(tid >> 5);          // H/4 == 32 chunks
    int c = (int)(tid & 31);
    int d = dst[e];
    if (c == 0) atomicAdd(&degf[d], 1.0f);
    const float4 v = ((const float4*)he)[(long long)e * (H / 4) + c];
    float* o = &sum_he[(long long)d * H + c * 4];
    atomicAdd(o + 0, v.x);
    atomicAdd(o + 1, v.y);
    atomicAdd(o + 2, v.z);
    atomicAdd(o + 3, v.w);
}

// mean_he = sum_he / max(deg,1);  deg_inv = 1/max(deg,1)
__global__ void finalize_mean_kernel(float* __restrict__ mean_he,
                                     float* __restrict__ deg_inv,
                                     const float* __restrict__ degf,
                                     int N) {
    long long tid = (long long)blockIdx.x * blockDim.x + threadIdx.x;
    if (tid >= (long long)N * H) return;
    int n = (int)(tid >> 7);
    int c = (int)(tid & (H - 1));
    float inv = 1.0f / fmaxf(degf[n], 1.0f);
    mean_he[tid] *= inv;
    if (c == 0) deg_inv[n] = inv;
}

// sum_hsrc -> mean_hsrc in place (scale rows by deg_inv)
__global__ void scale_hsrc_kernel(float* __restrict__ sum_hsrc,
                                  const float* __restrict__ deg_inv,
                                  int N) {
    long long tid = (long long)blockIdx.x * blockDim.x + threadIdx.x;
    if (tid >= (long long)N * H) return;
    sum_hsrc[tid] *= deg_inv[tid >> 7];
}

// One thread per (edge, float4 chunk): sum_hsrc[dst] += hv[src]
__global__ void scatter_kernel(const float* __restrict__ hv,
                               const int* __restrict__ src,
                               const int* __restrict__ dst,
                               float* __restrict__ sum_hsrc,
                               int E) {
    long long tid = (long long)blockIdx.x * blockDim.x + threadIdx.x;
    if (tid >= (long long)E * (H / 4)) return;
    int e = (int)(tid >> 5);
    int c = (int)(tid & 31);
    int s = src[e];
    int d = dst[e];
    const float4 v = ((const float4*)hv)[(long long)s * (H / 4) + c];
    float* o = &sum_hsrc[(long long)d * H + c * 4];
    atomicAdd(o + 0, v.x);
    atomicAdd(o + 1, v.y);
    atomicAdd(o + 2, v.z);
    atomicAdd(o + 3, v.w);
}

// --------------------------- TDM tile staging -------------------------------
// 1-D DMA: copy `nelem` contiguous f32 from global memory into LDS.
// Descriptor per CDNA5 ISA 8.3/8.4 (normal mode, no gather/iterate/pad).
// All fields are wave-uniform -> SGPR groups, as VIMAGE encoding requires.
__device__ __forceinline__ void tdm_load_f32(const float* gsrc,
                                             unsigned lds_byte_off,
                                             unsigned nelem) {
    unsigned long long ga = (unsigned long long)(uintptr_t)gsrc;
    v4u g0;
    g0.x = 1u;                                         // count=1, user descriptor
    g0.y = lds_byte_off;                               // lds_addr
    g0.z = (unsigned)(ga & 0xffffffffu);               // global_addr[31:0]
    g0.w = (unsigned)((ga >> 32) & 0x1ffffffu)         // global_addr[56:32]
         | (2u << 30);                                 // type=2 ("image")
    v8u g1;
    g1.s0 = (2u << 16);                                // data_size=2 -> 4 bytes
    g1.s1 = (nelem & 0xffffu) << 16;                   // tensor_dim0[15:0]
    g1.s2 = ((nelem >> 16) & 0xffffu)                  // tensor_dim0[31:16]
          | (1u << 16);                                // tensor_dim1 = 1
    g1.s3 = (nelem & 0xffffu) << 16;                   // tile_dim0 = nelem
    g1.s4 = 1u;                                        // tile_dim1 = 1
    g1.s5 = nelem;                                     // tensor_dim0_stride[31:0]
    g1.s6 = 0u;                                        // stride hi, dim1_stride lo
    g1.s7 = 0u;
    asm volatile("tensor_load_to_lds %0, %1" :: "s"(g0), "s"(g1) : "memory");
}

// ----------------------------- WMMA helpers --------------------------------
// Fragment layouts per CDNA5 ISA 7.12.2 (wave32):
//   A (16x4 f32):  lanes 0-15 -> row M=lane, K={0,1}; lanes 16-31 -> K={2,3}
//   B (4x16 f32):  lane group selects K half, lane%16 selects column N
//   C/D (16x16):   VGPR r -> M = r + 8*(lane/16), N = col0 + lane%16

__device__ __forceinline__ v8f tile_bias(const float* __restrict__ bias, int col0) {
    const int lane = threadIdx.x & (WAVE - 1);
    const float b = bias[col0 + (lane & 15)];
    v8f acc;
#pragma unroll
    for (int i = 0; i < 8; ++i) acc[i] = b;
    return acc;
}

// Two row-tiles against one shared B stream: each B fragment feeds 2 WMMAs.
__device__ __forceinline__ void gemm2(v8f& acc0, v8f& acc1,
                                      const float* a0, const float* a1,
                                      const float* __restrict__ brow, int K) {
#pragma unroll 4
    for (int k = 0; k < K; k += 4) {
        v2f bf  = *(const v2f*)(brow + k);
        v2f af0 = *(const v2f*)(a0 + k);
        v2f af1 = *(const v2f*)(a1 + k);
        acc0 = __builtin_amdgcn_wmma_f32_16x16x4_f32(
            false, af0, false, bf, (short)0, acc0, false, false);
        acc1 = __builtin_amdgcn_wmma_f32_16x16x4_f32(
            false, af1, false, bf, (short)0, acc1, false, false);
    }
}

__device__ __forceinline__ float sigmoidf(float x) {
    return 1.0f / (1.0f + __expf(-x));
}

// --------------------------- fused round kernel ----------------------------
// One wave handles MT=2 tiles of 16 node rows each. EXEC stays all-1s:
// tail tiles are clamped (duplicate compute) and only the store is guarded
// by a wave-uniform predicate.
__global__ __launch_bounds__(WAVE) void round_kernel(
    const float* __restrict__ hv_in,
    const float* __restrict__ mean_hsrc,  // pre-scaled by deg_inv
    const float* __restrict__ mean_he,
    const float* __restrict__ Wmsg,   // [2H, 3H] row-major, this round
    const float* __restrict__ bmsg,   // [2H]
    const float* __restrict__ Wih,    // [3H, 2H]
    const float* __restrict__ Whh,    // [3H, H]
    const float* __restrict__ bih,    // [3H]
    const float* __restrict__ bhh,    // [3H]
    float* __restrict__ hv_out,
    int N) {

    __shared__ float hv_lds[MT * 16 * H];      // 16 KB: old hv (A operand + blend)
    __shared__ float a_lds[MT * 16 * 2 * H];   // 32 KB: message activation

    const int lane = threadIdx.x & (WAVE - 1);
    const int half = lane >> 4;
    const int lrow = lane & 15;

    // Row bases for the two tiles (clamped for the tail; store-guarded).
    int b0 = (blockIdx.x * MT + 0) * 16;
    int b1 = (blockIdx.x * MT + 1) * 16;
    const bool v1 = (b1 < N);                  // tile 0 always valid
    if (b1 >= N) b1 = N - 16;                  // duplicate last tile's rows

    // ---- stage old hv into LDS via the Tensor Data Mover ----
    // Each 16-row tile of hv is one contiguous 16*H*4 = 8 KB block.
    {
        unsigned lds0 = (unsigned)(uintptr_t)&hv_lds[0];
        tdm_load_f32(hv_in + (long long)b0 * H, lds0, 16 * H);
        tdm_load_f32(hv_in + (long long)b1 * H, lds0 + 16 * H * 4, 16 * H);
        __builtin_amdgcn_s_wait_tensorcnt(0);
    }
    __syncthreads();

    // Per-lane A-fragment bases
    const float* ahv0 = &hv_lds[lrow * H + 2 * half];
    const float* ahv1 = &hv_lds[(16 + lrow) * H + 2 * half];
    const float* ams0 = mean_hsrc + (long long)(b0 + lrow) * H + 2 * half;
    const float* ams1 = mean_hsrc + (long long)(b1 + lrow) * H + 2 * half;
    const float* ame0 = mean_he + (long long)(b0 + lrow) * H + 2 * half;
    const float* ame1 = mean_he + (long long)(b1 + lrow) * H + 2 * half;
    const float* aa0 = &a_lds[lrow * 2 * H + 2 * half];
    const float* aa1 = &a_lds[(16 + lrow) * 2 * H + 2 * half];

    // ---- phase 1: a[.,256] = [hv|mean_hsrc|mean_he] @ Wmsg^T + bmsg ----
#pragma unroll 1
    for (int j = 0; j < 16; ++j) {
        const int col0 = j * 16;
        const float* brow = Wmsg + (long long)(col0 + lrow) * (3 * H) + 2 * half;
        v8f acc0 = tile_bias(bmsg, col0);
        v8f acc1 = acc0;
        gemm2(acc0, acc1, ahv0, ahv1, brow, H);            // k in [0,128)
        gemm2(acc0, acc1, ams0, ams1, brow + H, H);        // k in [128,256)
        gemm2(acc0, acc1, ame0, ame1, brow + 2 * H, H);    // k in [256,384)
#pragma unroll
        for (int r = 0; r < 8; ++r) {
            a_lds[(r + 8 * half) * 2 * H + col0 + lrow] = acc0[r];
            a_lds[(16 + r + 8 * half) * 2 * H + col0 + lrow] = acc1[r];
        }
    }
    __syncthreads();

    // ---- phase 2: GRU gates, one 16-wide H block at a time ----
    // r-tile (cols jh16), z-tile (cols 128+jh16), n-tile (cols 256+jh16)
    // share the (lane, component) -> (m, h) mapping: pure register math.
#pragma unroll 1
    for (int jh = 0; jh < 8; ++jh) {
        const int hc0 = jh * 16;
        v8f r0, r1, z0, z1;

        // reset gate: gi/gh columns hc0
        {
            const int col0 = hc0;
            v8f gi0 = tile_bias(bih, col0); v8f gi1 = gi0;
            gemm2(gi0, gi1, aa0, aa1, Wih + (long long)(col0 + lrow) * (2 * H) + 2 * half, 2 * H);
            v8f gh0 = tile_bias(bhh, col0); v8f gh1 = gh0;
            gemm2(gh0, gh1, ahv0, ahv1, Whh + (long long)(col0 + lrow) * H + 2 * half, H);
#pragma unroll
            for (int r = 0; r < 8; ++r) {
                r0[r] = sigmoidf(gi0[r] + gh0[r]);
                r1[r] = sigmoidf(gi1[r] + gh1[r]);
            }
        }
        // update gate: gi/gh columns H + hc0
        {
            const int col0 = H + hc0;
            v8f gi0 = tile_bias(bih, col0); v8f gi1 = gi0;
            gemm2(gi0, gi1, aa0, aa1, Wih + (long long)(col0 + lrow) * (2 * H) + 2 * half, 2 * H);
            v8f gh0 = tile_bias(bhh, col0); v8f gh1 = gh0;
            gemm2(gh0, gh1, ahv0, ahv1, Whh + (long long)(col0 + lrow) * H + 2 * half, H);
#pragma unroll
            for (int r = 0; r < 8; ++r) {
                z0[r] = sigmoidf(gi0[r] + gh0[r]);
                z1[r] = sigmoidf(gi1[r] + gh1[r]);
            }
        }
        // candidate + blend: gi/gh columns 2H + hc0
        {
            const int col0 = 2 * H + hc0;
            v8f gi0 = tile_bias(bih, col0); v8f gi1 = gi0;
            gemm2(gi0, gi1, aa0, aa1, Wih + (long long)(col0 + lrow) * (2 * H) + 2 * half, 2 * H);
            v8f gh0 = tile_bias(bhh, col0); v8f gh1 = gh0;
            gemm2(gh0, gh1, ahv0, ahv1, Whh + (long long)(col0 + lrow) * H + 2 * half, H);
#pragma unroll
            for (int r = 0; r < 8; ++r) {
                const int m = r + 8 * half;
                const int hc = hc0 + lrow;
                float n0 = tanhf(gi0[r] + r0[r] * gh0[r]);
                float n1 = tanhf(gi1[r] + r1[r] * gh1[r]);
                float h0 = hv_lds[m * H + hc];
                float h1 = hv_lds[(16 + m) * H + hc];
                hv_out[(long long)(b0 + m) * H + hc] = (1.0f - z0[r]) * n0 + z0[r] * h0;
                if (v1)
                    hv_out[(long long)(b1 + m) * H + hc] = (1.0f - z1[r]) * n1 + z1[r] * h1;
            }
        }
    }
}

// ------------------------------- launcher ----------------------------------

extern "C" void kernel_launch(void* const* d_in, const int* in_sizes, int n_in,
                              void* d_out, int out_size, void* d_ws, size_t ws_size,
                              hipStream_t stream) {
    const float* hv   = (const float*)d_in[0];   // [N,H]
    const float* he   = (const float*)d_in[1];   // [E,H]
    const int*   src  = (const int*)d_in[2];     // [E]
    const int*   dst  = (const int*)d_in[3];     // [E]
    const float* Wmsg = (const float*)d_in[4];   // [T,2H,3H]
    const float* bmsg = (const float*)d_in[5];   // [T,2H]
    const float* Wih  = (const float*)d_in[6];   // [T,3H,2H]
    const float* Whh  = (const float*)d_in[7];   // [T,3H,H]
    const float* bih  = (const float*)d_in[8];   // [T,3H]
    const float* bhh  = (const float*)d_in[9];   // [T,3H]

    const int N = in_sizes[0] / H;
    const int E = in_sizes[2];
    const int T = in_sizes[5] / (2 * H);

    // workspace layout
    float* ws       = (float*)d_ws;
    float* degf     = ws;                       // [N]
    float* deg_inv  = degf + N;                 // [N]
    float* mean_he  = deg_inv + N;              // [N,H]
    float* sum_hsrc = mean_he + (long long)N * H;   // [N,H] (becomes mean_hsrc)
    float* hv_tmp   = sum_hsrc + (long long)N * H;  // [N,H]

    const int TB = 256;
    long long nNH = (long long)N * H;
    long long nE4 = (long long)E * (H / 4);

    // degree + mean edge features (round-invariant)
    zero_kernel<<<(int)((N + TB - 1) / TB), TB, 0, stream>>>(degf, N);
    zero_kernel<<<(int)((nNH + TB - 1) / TB), TB, 0, stream>>>(mean_he, nNH);
    deg_he_kernel<<<(int)((nE4 + TB - 1) / TB), TB, 0, stream>>>(he, dst, degf, mean_he, E);
    finalize_mean_kernel<<<(int)((nNH + TB - 1) / TB), TB, 0, stream>>>(mean_he, deg_inv, degf, N);

    const int ntiles = (N + 15) / 16;
    const int nblocks = (ntiles + MT - 1) / MT;

    for (int t = 0; t < T; ++t) {
        const float* hin  = (t == 0) ? hv : hv_tmp;
        float*       hout = (t == T - 1) ? (float*)d_out : hv_tmp;

        zero_kernel<<<(int)((nNH + TB - 1) / TB), TB, 0, stream>>>(sum_hsrc, nNH);
        scatter_kernel<<<(int)((nE4 + TB - 1) / TB), TB, 0, stream>>>(hin, src, dst, sum_hsrc, E);
        scale_hsrc_kernel<<<(int)((nNH + TB - 1) / TB), TB, 0, stream>>>(sum_hsrc, deg_inv, N);

        round_kernel<<<nblocks, WAVE, 0, stream>>>(
            hin, sum_hsrc, mean_he,
            Wmsg + (long long)t * 2 * H * 3 * H,
            bmsg + (long long)t * 2 * H,
            Wih  + (long long)t * 3 * H * 2 * H,
            Whh  + (long long)t * 3 * H * H,
            bih  + (long long)t * 3 * H,
            bhh  + (long long)t * 3 * H,
            hout, N);
    }
}